// AdderModel_2018634629539
// MI455X (gfx1250) — compile-verified
//
#include <hip/hip_runtime.h>
#include <math.h>

// ---------------- problem constants ----------------
#define SEQ     4096
#define VOCAB   10
#define TPB     256
#define NCHUNK  (SEQ / TPB)    // 16 chunks per row
#define NRES    19             // rope period in integer steps
#define NPAIR   (VOCAB * NRES) // 190
#define NTAB    (VOCAB * NPAIR)// 1900 distinct (t_i, t_j, residue) weights

typedef __attribute__((ext_vector_type(2))) float v2f;
typedef __attribute__((ext_vector_type(8))) float v8f;

struct alignas(16) TabEntry { double e; double ev; };  // one ds_load_b128 per gather

// ------------------------------------------------------------------
// Kernel 1: build exp-weight table  T[t][d][r] = {E, E*val(d)} (double)
//   E = exp(ATTN_SCALE * g(t) * g(d) * cos(r*w - PHI)),  |logit| <= ~106
// ------------------------------------------------------------------
__device__ __forceinline__ float gv_of(int d, float cs) {
    float df = (float)d;
    float h0 = cs - df * df / cs;
    float h1 = -df;
    float rinv = rsqrtf((h0 * h0 + h1 * h1) * 0.5f + 1e-6f);
    float x0 = h0 * rinv;
    return x0 * rsqrtf(x0 * x0 * 0.5f + 1e-6f);    // g(u) ~ sign(u)*sqrt(2)
}

__device__ __forceinline__ float val_of(int d, float cs) {
    float df = (float)d;
    float h0 = cs - df * df / cs;
    float h1 = -df;
    float rinv = rsqrtf((h0 * h0 + h1 * h1) * 0.5f + 1e-6f);
    float x1 = h1 * rinv;
    const float V_FACTOR = -22.0f / 1.41421356237309515f;  // -22/sqrt(2)
    return x1 * V_FACTOR * cs;
}

__global__ void build_tables_kernel(const float* __restrict__ c,
                                    TabEntry* __restrict__ T) {
    int idx = blockIdx.x * blockDim.x + threadIdx.x;
    if (idx >= NTAB) return;
    const float cs = c[0];

    int t   = idx / NPAIR;
    int rem = idx % NPAIR;
    int d   = rem / NRES;
    int r   = rem % NRES;

    const double PI = 3.14159265358979323846;
    double omega = 2.0 * PI / 19.0;
    double phi   = omega * 10.3;                           // OMEGA*(10+PEAK_EPS)
    double amp   = log(10.0) / (cos(omega * 0.3) - cos(omega * 0.7));
    double scale = 0.5 * amp;                              // HEAD_DIM^-0.5 * QKNS^2

    double L = scale * (double)gv_of(t, cs) * (double)gv_of(d, cs)
                     * cos((double)r * omega - phi);
    double e = exp(L);                                     // fits f64 (e^106 ~ 1e46)
    T[idx].e  = e;
    T[idx].ev = e * (double)val_of(d, cs);
}

// ------------------------------------------------------------------
// Kernel 2: per-chunk (token, pos mod 19) histograms
//   H[row][chunk][d*19+m] = #{ j in chunk : t_j = d, j%19 = m }
// ------------------------------------------------------------------
__global__ void __launch_bounds__(TPB)
chunk_hist_kernel(const int* __restrict__ tokens, int* __restrict__ H) {
    __shared__ int hist[NPAIR];
    const int tid   = threadIdx.x;
    const int chunk = blockIdx.x;
    const int row   = blockIdx.y;

    if (tid < NPAIR) hist[tid] = 0;
    __syncthreads();

    int pos = chunk * TPB + tid;
    int t   = tokens[row * SEQ + pos];
    atomicAdd(&hist[t * NRES + pos % NRES], 1);
    __syncthreads();

    if (tid < NPAIR)
        H[(row * NCHUNK + chunk) * NPAIR + tid] = hist[tid];
}

// ------------------------------------------------------------------
// Kernel 3: attention (histogram prefix + exact in-chunk tail) + MLP
//           + WMMA f32 16x16x4 output projection
//   grid = (NCHUNK, batch); block = 256; 1 thread = 1 query position
// ------------------------------------------------------------------
__global__ void __launch_bounds__(TPB)
attn_mlp_proj_kernel(const int* __restrict__ tokens,
                     const float* __restrict__ c,
                     const TabEntry* __restrict__ Tg,
                     const int* __restrict__ Hg,
                     float* __restrict__ out) {
    __shared__ TabEntry T[NTAB];      // 30.4 KB weight table
    __shared__ double   cnt[NPAIR];   // prefix histogram (as f64 for pure FMA)
    __shared__ int      toks19[TPB];  // token*19 for this block's chunk
    __shared__ float    ov0[TPB];     // final out_vec component 0
    __shared__ float    ov1[TPB];     // final out_vec component 1

    const int tid     = threadIdx.x;
    const int k       = blockIdx.x;   // chunk index
    const int row     = blockIdx.y;
    const int rowBase = row * SEQ;
    const int i       = k * TPB + tid;
    const float cs    = c[0];

    // stage weight table into LDS (b128 copies)
    for (int e = tid; e < NTAB; e += TPB) T[e] = Tg[e];

    // prefix histogram = sum of chunk histograms strictly before this chunk
    for (int e = tid; e < NPAIR; e += TPB) {
        const int* Hrow = Hg + row * NCHUNK * NPAIR;
        int s = 0;
        for (int cc = 0; cc < k; ++cc) s += Hrow[cc * NPAIR + e];
        cnt[e] = (double)s;
    }

    const int ti = tokens[rowBase + i];
    toks19[tid] = NRES * ti;
    __syncthreads();

    const TabEntry* Tb = T + ti * NPAIR;
    double numer = 0.0, denom = 0.0;

    // ---- all full chunks before ours: 190 f64 FMAs off the histogram ----
    {
        const int im = i % NRES;
        int e = 0;
        for (int d = 0; d < VOCAB; ++d) {
            int rr = im;                      // (i - j) mod 19 for j%19 == m
            const TabEntry* Td = Tb + d * NRES;
            #pragma unroll
            for (int m = 0; m < NRES; ++m) {
                double cc = cnt[e++];         // uniform LDS broadcast
                TabEntry w = Td[rr];          // one ds_load_b128
                denom = fma(cc, w.e,  denom);
                numer = fma(cc, w.ev, numer);
                rr = (rr == 0) ? (NRES - 1) : (rr - 1);
            }
        }
    }

    // ---- exact causal tail inside our own chunk: j = k*TPB .. i ----
    {
        int r = tid % NRES;                   // (i - j) mod 19 at j = chunk start
        for (int jj = 0; jj <= tid; ++jj) {
            TabEntry w = Tb[toks19[jj] + r];  // one ds_load_b128
            denom += w.e;
            numer += w.ev;
            r = (r == 0) ? (NRES - 1) : (r - 1);
        }
    }
    float o0 = (float)(numer / denom);        // softmax-weighted value sum

    // ---- residual + MLP + final norm ----
    float tf = (float)ti;
    float h0 = cs - tf * tf / cs;             // tab[t][0]
    float h1 = -tf + o0;                      // tab[t][1] + attn_out

    float rinv = rsqrtf((h0 * h0 + h1 * h1) * 0.5f + 1e-6f);
    float hn0 = h0 * rinv, hn1 = h1 * rinv;

    float a  = -12.032f * cs;                 // ALPHA * cs
    float gc = 128.0f * cs;
    float g0 = hn0 * a + hn1 * gc;
    float g1 = hn0 * (a - gc / cs) + hn1 * gc;
    float up = hn0;
    float mix0 = (g0 / (1.0f + expf(-g0))) * up;    // silu(g)*up
    float mix1 = (g1 / (1.0f + expf(-g1))) * up;
    float y1 = (100.0f / 256.0f) * (mix1 - mix0);
    h1 += y1;

    rinv = rsqrtf((h0 * h0 + h1 * h1) * 0.5f + 1e-6f);
    const float inv_s2 = 0.70710678118654752f;      // 1/sqrt(MODEL_DIM)
    float nw0 = 0.1f * cs * inv_s2;
    float nw1 = -cs * 0.02f * inv_s2;               // -cs/(50*sqrt(2))
    ov0[tid] = h0 * rinv * nw0;
    ov1[tid] = h1 * rinv * nw1;
    __syncthreads();

    // ---- WMMA epilogue: out_vec(16x2) @ tab.T(2x10) via 16x16x4 f32 ----
    const int lane = tid & 31;
    const int wave = tid >> 5;

    // B 4x16: rows K=0,1 = tab columns (K=2,3 and N>=10 zero)
    v2f bM;
    float nf = (float)(lane & 15);
    bM.x = (lane < VOCAB) ? (cs - nf * nf / cs) : 0.0f;
    bM.y = (lane < VOCAB) ? (-nf)               : 0.0f;

    float* outRow = out + (size_t)(rowBase + k * TPB) * VOCAB;

    #pragma unroll
    for (int t2 = 0; t2 < 2; ++t2) {
        int tileIdx = wave * 2 + t2;          // 16 position-tiles per block
        // A 16x4: lanes 0..15 carry M=0..15; VGPR0=K0 (ov0), VGPR1=K1 (ov1)
        v2f aM;
        aM.x = (lane < 16) ? ov0[tileIdx * 16 + lane] : 0.0f;
        aM.y = (lane < 16) ? ov1[tileIdx * 16 + lane] : 0.0f;

        v8f acc = {};
        acc = __builtin_amdgcn_wmma_f32_16x16x4_f32(
                  /*neg_a=*/false, aM, /*neg_b=*/false, bM,
                  /*c_mod=*/(short)0, acc, /*reuse_a=*/false, /*reuse_b=*/false);

        // D layout: VGPR v of lane l holds (M = v + 8*(l>=16), N = l&15)
        int N     = lane & 15;
        int mbase = tileIdx * 16 + ((lane >= 16) ? 8 : 0);
        if (N < VOCAB) {
            #pragma unroll
            for (int v = 0; v < 8; ++v)
                outRow[(size_t)(mbase + v) * VOCAB + N] = acc[v];
        }
    }
}

// ------------------------------------------------------------------
extern "C" void kernel_launch(void* const* d_in, const int* in_sizes, int n_in,
                              void* d_out, int out_size, void* d_ws, size_t ws_size,
                              hipStream_t stream) {
    const int*   tokens = (const int*)d_in[0];
    const float* c      = (const float*)d_in[1];
    float*       out    = (float*)d_out;

    const int batch = in_sizes[0] / SEQ;     // 4

    TabEntry* T = (TabEntry*)d_ws;                      // 1900 * 16B = 30.4 KB
    int*      H = (int*)(T + NTAB);                     // batch*16*190 ints

    build_tables_kernel<<<dim3((NTAB + TPB - 1) / TPB), dim3(TPB), 0, stream>>>(c, T);

    dim3 grid(NCHUNK, batch);                // (16, batch)
    chunk_hist_kernel<<<grid, dim3(TPB), 0, stream>>>(tokens, H);
    attn_mlp_proj_kernel<<<grid, dim3(TPB), 0, stream>>>(tokens, c, T, H, out);
}